// TestGNN_62972810494186
// MI455X (gfx1250) — compile-verified
//
#include <hip/hip_runtime.h>
#include <hip/hip_bf16.h>
#include <math.h>

#define LN_EPS 1e-5f

typedef float v2f __attribute__((ext_vector_type(2)));
typedef float v8f __attribute__((ext_vector_type(8)));

// ---------------- degree / normalization ----------------
__global__ void k_deg_init(float* deg, int n) {
    int i = blockIdx.x * blockDim.x + threadIdx.x;
    if (i < n) deg[i] = 1.0f;  // self-loop contributes 1
}

__global__ void k_deg_accum(const int* __restrict__ dst, float* deg, int E) {
    int e = blockIdx.x * blockDim.x + threadIdx.x;
    if (e < E) atomicAdd(&deg[dst[e]], 1.0f);
}

__global__ void k_deg_rsqrt(float* deg, int n) {
    int i = blockIdx.x * blockDim.x + threadIdx.x;
    if (i < n) deg[i] = rsqrtf(deg[i]);
}

// ---------------- fp32 WMMA GEMM: Y[N,FOUT] = X[N,K] @ W[K,FOUT] (+bias) ----
// Block = 4 waves; each wave computes one 16x16 output tile of column-tile tn.
// W tile (K x 16, zero-padded past FOUT) staged once in LDS, shared by all 4
// waves -> inner loop has no guards, no 64-bit address math, B comes from DS.
// A 16x4 layout : lanes 0-15 -> M=lane; half selects K{k,k+1} / K{k+2,k+3}
// B 4x16 layout : VGPR0 = rows {k, k+2}, VGPR1 = rows {k+1, k+3} by lane-half
// C/D 16x16     : VGPR r -> M = r + 8*half, N = lane&15
template <int K, int FOUT>
__global__ __launch_bounds__(128) void k_gemm_wmma(
        const float* __restrict__ X, const float* __restrict__ W,
        const float* __restrict__ bias, float* __restrict__ Y, int MT) {
    __shared__ float sW[K * 16];
    const int tn  = blockIdx.y;
    const int tid = threadIdx.y * 32 + threadIdx.x;
    // cooperative zero-padded fill of the K x 16 W tile
    for (int idx = tid; idx < K * 16; idx += 128) {
        const int kk  = idx >> 4;
        const int c   = idx & 15;
        const int col = tn * 16 + c;
        sW[idx] = (FOUT % 16 == 0 || col < FOUT) ? W[kk * FOUT + col] : 0.0f;
    }
    __syncthreads();

    const int tm = blockIdx.x * 4 + threadIdx.y;  // wave-uniform
    if (tm >= MT) return;                          // uniform exit, EXEC intact
    const int lane = threadIdx.x;
    const int half = lane >> 4;
    const int l16  = lane & 15;
    const int row  = tm * 16 + l16;

    v8f acc = {};
    const float* xr = X + (size_t)row * K + half * 2;
#pragma unroll 8
    for (int k = 0; k < K; k += 4) {
        v2f a, b;
        a.x = xr[k];
        a.y = xr[k + 1];
        const int kb = k + half * 2;
        b.x = sW[kb * 16 + l16];
        b.y = sW[(kb + 1) * 16 + l16];
        acc = __builtin_amdgcn_wmma_f32_16x16x4_f32(
            /*neg_a=*/false, a, /*neg_b=*/false, b,
            /*c_mod=*/(short)0, acc, /*reuse_a=*/false, /*reuse_b=*/false);
    }

    const int colD = tn * 16 + l16;
    if (FOUT % 16 != 0 && colD >= FOUT) return;    // only exists for FOUT=40
    const float bv = bias ? bias[colD] : 0.0f;
#pragma unroll
    for (int r = 0; r < 8; ++r) {
        const int m = tm * 16 + half * 8 + r;
        Y[(size_t)m * FOUT + colD] = acc[r] + bv;
    }
}

// ---------------- self-loop term: out[i,f] = h[i,f] * dinv[i]^2 -------------
__global__ void k_selfloop(const float* __restrict__ h, const float* __restrict__ dinv,
                           float* __restrict__ out, int n, int logF) {
    int idx = blockIdx.x * blockDim.x + threadIdx.x;
    if (idx >= (n << logF)) return;
    int i = idx >> logF;
    float d = dinv[i];
    out[idx] = h[idx] * d * d;
}

// ---------------- edge scatter: out[dst] += h[src]*dinv[src]*dinv[dst] ------
__global__ void k_scatter(const float* __restrict__ h, const int* __restrict__ src,
                          const int* __restrict__ dst, const float* __restrict__ dinv,
                          float* __restrict__ out, int E, int F4) {
    long long tid = (long long)blockIdx.x * blockDim.x + threadIdx.x;
    if (tid >= (long long)E * F4) return;
    int e = (int)(tid / F4);
    int c = (int)(tid % F4);
    int s = src[e], d = dst[e];
    float nrm = dinv[s] * dinv[d];
    float4 v = ((const float4*)h)[(size_t)s * F4 + c];
    float* o = out + ((size_t)d * F4 + c) * 4;
    atomicAdd(o + 0, v.x * nrm);
    atomicAdd(o + 1, v.y * nrm);
    atomicAdd(o + 2, v.z * nrm);
    atomicAdd(o + 3, v.w * nrm);
}

// ---------------- fused bias + ReLU + LayerNorm (one wave32 per node) -------
__global__ void k_bias_relu_ln(const float* __restrict__ conv, const float* __restrict__ bias,
                               const float* __restrict__ g, const float* __restrict__ beta,
                               float* __restrict__ out, int n, int F) {
    int warp = threadIdx.x >> 5;
    int lane = threadIdx.x & 31;
    int node = blockIdx.x * (blockDim.x >> 5) + warp;
    if (node >= n) return;
    const int cnt = F >> 5;  // 1 (F=32) or 2 (F=64)
    float vals[2];
    float sum = 0.0f;
    for (int j = 0; j < cnt; ++j) {
        int f = lane + j * 32;
        float v = conv[(size_t)node * F + f] + bias[f];
        v = fmaxf(v, 0.0f);
        vals[j] = v;
        sum += v;
    }
    for (int off = 16; off >= 1; off >>= 1) sum += __shfl_xor(sum, off, 32);
    float mu = sum / (float)F;
    float vs = 0.0f;
    for (int j = 0; j < cnt; ++j) { float dv = vals[j] - mu; vs += dv * dv; }
    for (int off = 16; off >= 1; off >>= 1) vs += __shfl_xor(vs, off, 32);
    float inv = rsqrtf(vs / (float)F + LN_EPS);
    for (int j = 0; j < cnt; ++j) {
        int f = lane + j * 32;
        out[(size_t)node * F + f] = (vals[j] - mu) * inv * g[f] + beta[f];
    }
}

// ---------------- emb = conv3 + b3 -> d_out ; relu(emb) -> scratch ----------
__global__ void k_emb_relu(const float* __restrict__ conv, const float* __restrict__ b3,
                           float* __restrict__ emb, float* __restrict__ relu_out, int n) {
    int idx = blockIdx.x * blockDim.x + threadIdx.x;
    if (idx >= n * 128) return;
    int f = idx & 127;
    float e = conv[idx] + b3[f];
    emb[idx] = e;
    relu_out[idx] = fmaxf(e, 0.0f);
}

// ---------------- log_softmax over C=40 columns, in place (wave per node) ---
__global__ void k_log_softmax(float* __restrict__ x, int n, int C) {
    int warp = threadIdx.x >> 5;
    int lane = threadIdx.x & 31;
    int node = blockIdx.x * (blockDim.x >> 5) + warp;
    if (node >= n) return;
    float vals[2];
    float mx = -3.402823466e+38f;
    int k = 0;
    for (int f = lane; f < C; f += 32, ++k) {
        vals[k] = x[(size_t)node * C + f];
        mx = fmaxf(mx, vals[k]);
    }
    for (int off = 16; off >= 1; off >>= 1) mx = fmaxf(mx, __shfl_xor(mx, off, 32));
    float s = 0.0f;
    for (int j = 0; j < k; ++j) s += __expf(vals[j] - mx);
    for (int off = 16; off >= 1; off >>= 1) s += __shfl_xor(s, off, 32);
    float ls = __logf(s);
    k = 0;
    for (int f = lane; f < C; f += 32, ++k)
        x[(size_t)node * C + f] = vals[k] - mx - ls;
}

static inline int cdiv(long long a, long long b) { return (int)((a + b - 1) / b); }

extern "C" void kernel_launch(void* const* d_in, const int* in_sizes, int n_in,
                              void* d_out, int out_size, void* d_ws, size_t ws_size,
                              hipStream_t stream) {
    const float* x    = (const float*)d_in[0];
    const int*   ei   = (const int*)d_in[1];
    const float* W1   = (const float*)d_in[2];
    const float* b1   = (const float*)d_in[3];
    const float* W2   = (const float*)d_in[4];
    const float* b2   = (const float*)d_in[5];
    const float* W3   = (const float*)d_in[6];
    const float* b3   = (const float*)d_in[7];
    const float* ln1g = (const float*)d_in[8];
    const float* ln1b = (const float*)d_in[9];
    const float* ln2g = (const float*)d_in[10];
    const float* ln2b = (const float*)d_in[11];
    const float* mW1  = (const float*)d_in[12];
    const float* mb1  = (const float*)d_in[13];
    const float* mW2  = (const float*)d_in[14];
    const float* mb2  = (const float*)d_in[15];

    const int N = in_sizes[0] / 128;   // 100000 (multiple of 16)
    const int E = in_sizes[1] / 2;     // 1600000
    const int* src = ei;
    const int* dst = ei + E;

    float* ws   = (float*)d_ws;
    float* dinv = ws;                               // [N]
    float* bufA = ws + N;                           // [N*128]
    float* bufB = bufA + (size_t)N * 128;           // [N*128]
    float* emb    = (float*)d_out;                  // [N*128]
    float* logits = emb + (size_t)N * 128;          // [N*40]

    const int TPB = 256;
    // gcn_norm: deg (with self loops) -> dinv
    k_deg_init <<<cdiv(N, TPB), TPB, 0, stream>>>(dinv, N);
    k_deg_accum<<<cdiv(E, TPB), TPB, 0, stream>>>(dst, dinv, E);
    k_deg_rsqrt<<<cdiv(N, TPB), TPB, 0, stream>>>(dinv, N);

    const int MT = N / 16;             // 6250 row tiles
    dim3 gblk(32, 4);                  // 4 waves per workgroup
    const int GX = cdiv(MT, 4);

    // ---- layer 1: F=32 ----
    k_gemm_wmma<128, 32><<<dim3(GX, 2), gblk, 0, stream>>>(x, W1, nullptr, bufA, MT);
    k_selfloop <<<cdiv((long long)N * 32, TPB), TPB, 0, stream>>>(bufA, dinv, bufB, N, 5);
    k_scatter  <<<cdiv((long long)E * 8, TPB), TPB, 0, stream>>>(bufA, src, dst, dinv, bufB, E, 8);
    k_bias_relu_ln<<<cdiv(N, 8), TPB, 0, stream>>>(bufB, b1, ln1g, ln1b, bufA, N, 32);

    // ---- layer 2: F=64 ----
    k_gemm_wmma<32, 64><<<dim3(GX, 4), gblk, 0, stream>>>(bufA, W2, nullptr, bufB, MT);
    k_selfloop <<<cdiv((long long)N * 64, TPB), TPB, 0, stream>>>(bufB, dinv, bufA, N, 6);
    k_scatter  <<<cdiv((long long)E * 16, TPB), TPB, 0, stream>>>(bufB, src, dst, dinv, bufA, E, 16);
    k_bias_relu_ln<<<cdiv(N, 8), TPB, 0, stream>>>(bufA, b2, ln2g, ln2b, bufB, N, 64);

    // ---- layer 3: F=128 ----
    k_gemm_wmma<64, 128><<<dim3(GX, 8), gblk, 0, stream>>>(bufB, W3, nullptr, bufA, MT);
    k_selfloop <<<cdiv((long long)N * 128, TPB), TPB, 0, stream>>>(bufA, dinv, bufB, N, 7);
    k_scatter  <<<cdiv((long long)E * 32, TPB), TPB, 0, stream>>>(bufA, src, dst, dinv, bufB, E, 32);
    k_emb_relu <<<cdiv((long long)N * 128, TPB), TPB, 0, stream>>>(bufB, b3, emb, bufA, N);

    // ---- MLP head + log_softmax ----
    k_gemm_wmma<128, 64><<<dim3(GX, 4), gblk, 0, stream>>>(bufA, mW1, mb1, bufB, MT);
    k_gemm_wmma<64, 40><<<dim3(GX, 3), gblk, 0, stream>>>(bufB, mW2, mb2, logits, MT);
    k_log_softmax<<<cdiv(N, 8), TPB, 0, stream>>>(logits, N, 40);
}